// RowParallelLinearSparse_89206470738366
// MI455X (gfx1250) — compile-verified
//
#include <hip/hip_runtime.h>

// ---------------------------------------------------------------------------
// RowParallelLinearSparse on MI455X (gfx1250, wave32).
// Reference weight is 2:4 structured sparse -> V_SWMMAC_F32_16X16X64_BF16
// (A compressed to half size, K=64 per op).
//   GEMM: M = 16384 (S*B), N = O = 2048, K = H = 2048.
// Pipeline: (1) pack weight to 2:4 compressed bf16 + 2-bit indices,
//           (2) convert x to bf16,
//           (3) double-buffered SWMMAC GEMM with async global->LDS staging.
// ---------------------------------------------------------------------------

typedef __attribute__((ext_vector_type(16))) __bf16     v16bf;
typedef __attribute__((ext_vector_type(32))) __bf16     v32bf;
typedef __attribute__((ext_vector_type(8)))  float      v8f;
typedef __attribute__((ext_vector_type(4)))  float      f32x4;
typedef __attribute__((ext_vector_type(4)))  unsigned   u32x4;
typedef __attribute__((ext_vector_type(4)))  int        i32x4;

#define DIM_H 2048
#define DIM_O 2048
#define DIM_M 16384   // S*B = 8192*2

// --- CDNA5 async global->LDS feature probes --------------------------------
#if defined(__has_builtin)
#if __has_builtin(__builtin_amdgcn_global_load_async_to_lds_b128)
#define HAVE_ASYNC_B128 1
#endif
#if __has_builtin(__builtin_amdgcn_global_load_async_to_lds_b32)
#define HAVE_ASYNC_B32 1
#endif
#if __has_builtin(__builtin_amdgcn_s_wait_asynccnt)
#define HAVE_WAIT_ASYNC 1
#endif
#endif
#ifndef HAVE_ASYNC_B128
#define HAVE_ASYNC_B128 0
#endif
#ifndef HAVE_ASYNC_B32
#define HAVE_ASYNC_B32 0
#endif
#ifndef HAVE_WAIT_ASYNC
#define HAVE_WAIT_ASYNC 0
#endif

#define GAS(T) __attribute__((address_space(1))) T
#define LAS(T) __attribute__((address_space(3))) T

// 16-byte global -> LDS copy (async, ASYNCcnt-tracked, no VGPR data)
__device__ __forceinline__ void copy16_g2l(const void* g, void* l) {
#if HAVE_ASYNC_B128
  __builtin_amdgcn_global_load_async_to_lds_b128(
      (GAS(i32x4)*)g, (LAS(i32x4)*)l, 0, 0);
#else
  *(u32x4*)l = *(const u32x4*)g;
#endif
}
// 4-byte global -> LDS copy
__device__ __forceinline__ void copy4_g2l(const void* g, void* l) {
#if HAVE_ASYNC_B32
  __builtin_amdgcn_global_load_async_to_lds_b32(
      (GAS(int)*)g, (LAS(int)*)l, 0, 0);
#else
  *(unsigned*)l = *(const unsigned*)g;
#endif
}
__device__ __forceinline__ void wait_async_done() {
#if HAVE_WAIT_ASYNC
  __builtin_amdgcn_s_wait_asynccnt(0);
#endif
}

// float -> bf16, round to nearest even
__device__ __forceinline__ unsigned short f2bf(float f) {
  unsigned u = __float_as_uint(f);
  u = (u + 0x7FFFu + ((u >> 16) & 1u)) >> 16;
  return (unsigned short)u;
}

// ---------------------------------------------------------------------------
// Kernel 1: 2:4 pack. Per group of 4 along H: reference keeps rank>=2 entries
// of the stable ascending argsort of |w|. Emit the two kept values (index
// order i0<i1) as bf16 into w_pack [O][H/2], and the 4-bit code (i0 | i1<<2)
// into w_idx [O][H/32] (group g at bits 4*(g%8)) per ISA 7.12.4 index layout.
// One thread = 8 groups = 32 H-elements.
// ---------------------------------------------------------------------------
__global__ __launch_bounds__(256) void nm_pack_w(
    const float* __restrict__ w,
    unsigned short* __restrict__ wpack,   // [O][H/2] bf16
    unsigned* __restrict__ widx) {        // [O][H/32] u32
  size_t t = (size_t)blockIdx.x * blockDim.x + threadIdx.x;
  const f32x4* src = (const f32x4*)w + t * 8;
  unsigned short pk[16];
  unsigned idxword = 0;
#pragma unroll
  for (int g = 0; g < 8; ++g) {
    f32x4 v = src[g];
    float vv[4] = {v.x, v.y, v.z, v.w};
    float av[4] = {fabsf(v.x), fabsf(v.y), fabsf(v.z), fabsf(v.w)};
    int i0 = -1, i1 = -1;
#pragma unroll
    for (int j = 0; j < 4; ++j) {
      int rank = 0;
#pragma unroll
      for (int i = 0; i < 4; ++i)
        rank += ((av[i] < av[j]) || (av[i] == av[j] && i < j)) ? 1 : 0;
      if (rank >= 2) { if (i0 < 0) i0 = j; else i1 = j; }   // kept, i0 < i1
    }
    pk[g * 2]     = f2bf(vv[i0]);
    pk[g * 2 + 1] = f2bf(vv[i1]);
    idxword |= ((unsigned)(i0 | (i1 << 2))) << (4 * g);
  }
  u32x4 o0, o1;
  o0.x = pk[0] | ((unsigned)pk[1] << 16);   o0.y = pk[2] | ((unsigned)pk[3] << 16);
  o0.z = pk[4] | ((unsigned)pk[5] << 16);   o0.w = pk[6] | ((unsigned)pk[7] << 16);
  o1.x = pk[8] | ((unsigned)pk[9] << 16);   o1.y = pk[10] | ((unsigned)pk[11] << 16);
  o1.z = pk[12] | ((unsigned)pk[13] << 16); o1.w = pk[14] | ((unsigned)pk[15] << 16);
  u32x4* dst = (u32x4*)(wpack + t * 16);
  dst[0] = o0;
  dst[1] = o1;
  widx[t] = idxword;
}

// ---------------------------------------------------------------------------
// Kernel 2: convert x to bf16 (8 elements / thread).
// ---------------------------------------------------------------------------
__global__ __launch_bounds__(256) void convert_x(
    const float* __restrict__ x, unsigned short* __restrict__ xbf) {
  size_t t = (size_t)blockIdx.x * blockDim.x + threadIdx.x;
  const f32x4* src = (const f32x4*)x + t * 2;
  f32x4 a = src[0], b = src[1];
  u32x4 o;
  o.x = (unsigned)f2bf(a.x) | ((unsigned)f2bf(a.y) << 16);
  o.y = (unsigned)f2bf(a.z) | ((unsigned)f2bf(a.w) << 16);
  o.z = (unsigned)f2bf(b.x) | ((unsigned)f2bf(b.y) << 16);
  o.w = (unsigned)f2bf(b.z) | ((unsigned)f2bf(b.w) << 16);
  *((u32x4*)xbf + t) = o;
}

// ---------------------------------------------------------------------------
// Kernel 3: sparse SWMMAC GEMM, double-buffered async staging.
//   WG = 256 threads = 8 wave32s; 128 (o) x 128 (m) output tile per WG.
//   Wave grid 4 (o x32) x 2 (m x64): 2x4 = 8 D-tiles/wave.
// ---------------------------------------------------------------------------
#define XT (128 * 64)   // x tile: bf16
#define WT (128 * 32)   // compressed w tile: bf16
#define IT (128 * 2)    // index tile: u32

__device__ __forceinline__ void stage_tiles(
    int tid, int mBase, int oBase, int k0,
    const unsigned short* __restrict__ xbf,
    const unsigned short* __restrict__ wpack,
    const unsigned* __restrict__ widx,
    unsigned short* xs, unsigned short* wp, unsigned* wi) {
  // x tile: 128 x 64 bf16 = 1024 x 16B
#pragma unroll
  for (int it = 0; it < 4; ++it) {
    int c = tid + 256 * it;
    int r = c >> 3, col = (c & 7) * 8;
    copy16_g2l(&xbf[(size_t)(mBase + r) * DIM_H + k0 + col], &xs[r * 64 + col]);
  }
  // compressed w tile: 128 x 32 bf16 = 512 x 16B
#pragma unroll
  for (int it = 0; it < 2; ++it) {
    int c = tid + 256 * it;
    int r = c >> 2, col = (c & 3) * 8;
    copy16_g2l(&wpack[(size_t)(oBase + r) * (DIM_H / 2) + k0 / 2 + col],
               &wp[r * 32 + col]);
  }
  // sparse indices: 128 rows x 2 u32
  {
    int r = tid >> 1, j = tid & 1;
    copy4_g2l(&widx[(size_t)(oBase + r) * (DIM_H / 32) + k0 / 32 + j],
              &wi[r * 2 + j]);
  }
}

__global__ __launch_bounds__(256) void gemm_bf16_swmmac(
    const unsigned short* __restrict__ xbf,     // [M][H] bf16
    const unsigned short* __restrict__ wpack,   // [O][H/2] bf16 compressed
    const unsigned* __restrict__ widx,          // [O][H/32] sparse indices
    const float* __restrict__ bias,             // [O]
    float* __restrict__ out) {                  // [M][O] fp32
  __shared__ __align__(16) unsigned short xs[2][XT];   // 2 x 16 KB
  __shared__ __align__(16) unsigned short wp[2][WT];   // 2 x  8 KB
  __shared__ unsigned wi[2][IT];                       // 2 x  1 KB

  const int tid  = threadIdx.x;
  const int lane = tid & 31;
  const int wave = tid >> 5;
  const int wo   = wave & 3;    // o-block of 32
  const int wm   = wave >> 2;   // m-block of 64
  const int oBase = blockIdx.x * 128;
  const int mBase = blockIdx.y * 128;

  const int row  = lane & 15;
  const int half = lane >> 4;

  v8f acc[2][4];
#pragma unroll
  for (int oi = 0; oi < 2; ++oi)
#pragma unroll
    for (int mi = 0; mi < 4; ++mi)
#pragma unroll
      for (int e = 0; e < 8; ++e) acc[oi][mi][e] = 0.0f;

  // Prologue: stage tile 0 into buffer 0.
  stage_tiles(tid, mBase, oBase, 0, xbf, wpack, widx, xs[0], wp[0], wi[0]);
  wait_async_done();
  __syncthreads();

  int buf = 0;
  for (int k0 = 0; k0 < DIM_H; k0 += 64) {    // expanded K step = 64
    // Issue async staging of the NEXT tile into the other buffer, then
    // compute on the current one -> global->LDS overlaps the SWMMACs.
    if (k0 + 64 < DIM_H) {
      stage_tiles(tid, mBase, oBase, k0 + 64, xbf, wpack, widx,
                  xs[buf ^ 1], wp[buf ^ 1], wi[buf ^ 1]);
      __builtin_prefetch(&xbf[(size_t)(mBase + (tid >> 1)) * DIM_H + k0 + 128], 0, 1);
    }

    // ---- A fragments: compressed 16x32 bf16, dense-A VGPR layout.
    // lanes 0-15: cK {0..7} U {16..23}; lanes 16-31: cK {8..15} U {24..31}.
    v16bf afrag[2];
    unsigned aidx[2];
#pragma unroll
    for (int oi = 0; oi < 2; ++oi) {
      int r0 = wo * 32 + oi * 16 + row;
      int kb = half * 8;
      union { u32x4 q[2]; v16bf v; } ua;
      ua.q[0] = *(const u32x4*)&wp[buf][r0 * 32 + kb];
      ua.q[1] = *(const u32x4*)&wp[buf][r0 * 32 + kb + 16];
      afrag[oi] = ua.v;
      aidx[oi]  = wi[buf][r0 * 2 + half];  // lane<16: K 0..31; lane>=16: K 32..63
    }
    // ---- B fragments: 64x16 bf16 dense. Per ISA 7.12.4:
    // V0..7 lanes 0-15 K=0..15 / lanes 16-31 K=16..31; V8..15: +32.
    v32bf bfrag[4];
#pragma unroll
    for (int mi = 0; mi < 4; ++mi) {
      int r0 = wm * 64 + mi * 16 + row;
      int kb = half * 16;
      union { u32x4 q[4]; v32bf v; } ub;
      ub.q[0] = *(const u32x4*)&xs[buf][r0 * 64 + kb];
      ub.q[1] = *(const u32x4*)&xs[buf][r0 * 64 + kb + 8];
      ub.q[2] = *(const u32x4*)&xs[buf][r0 * 64 + kb + 32];
      ub.q[3] = *(const u32x4*)&xs[buf][r0 * 64 + kb + 40];
      bfrag[mi] = ub.v;
    }
#pragma unroll
    for (int oi = 0; oi < 2; ++oi)
#pragma unroll
      for (int mi = 0; mi < 4; ++mi)
        acc[oi][mi] = __builtin_amdgcn_swmmac_f32_16x16x64_bf16(
            false, afrag[oi], false, bfrag[mi], acc[oi][mi],
            (int)aidx[oi], false, false);

    // Next buffer must have landed; all waves done reading current buffer.
    wait_async_done();
    __syncthreads();
    buf ^= 1;
  }

  // Store: D element e in lane l -> (o = otile + (l<16?0:8) + e, m = mtile + l%16)
  const int rbase = half * 8;
#pragma unroll
  for (int oi = 0; oi < 2; ++oi) {
    int oCol = oBase + wo * 32 + oi * 16 + rbase;
    f32x4 b0 = *(const f32x4*)&bias[oCol];
    f32x4 b1 = *(const f32x4*)&bias[oCol + 4];
#pragma unroll
    for (int mi = 0; mi < 4; ++mi) {
      int m = mBase + wm * 64 + mi * 16 + row;
      float* dst = &out[(size_t)m * DIM_O + oCol];
      f32x4 lo, hi;
#pragma unroll
      for (int e = 0; e < 4; ++e) {
        lo[e] = acc[oi][mi][e] + b0[e];
        hi[e] = acc[oi][mi][4 + e] + b1[e];
      }
      *(f32x4*)dst       = lo;
      *(f32x4*)(dst + 4) = hi;
    }
  }
}

// ---------------------------------------------------------------------------
extern "C" void kernel_launch(void* const* d_in, const int* in_sizes, int n_in,
                              void* d_out, int out_size, void* d_ws, size_t ws_size,
                              hipStream_t stream) {
  (void)in_sizes; (void)n_in; (void)out_size; (void)ws_size;
  const float* x    = (const float*)d_in[0];   // [8192, 2, 2048]
  const float* w    = (const float*)d_in[1];   // [2048, 2048]
  const float* bias = (const float*)d_in[2];   // [2048]
  float* out        = (float*)d_out;           // [16384, 2048]

  char* ws = (char*)d_ws;
  unsigned short* wpack = (unsigned short*)ws;                      // 4 MB
  unsigned*       widx  = (unsigned*)(ws + 4u * 1024 * 1024);       // 512 KB
  unsigned short* xbf   = (unsigned short*)(ws + 8u * 1024 * 1024); // 64 MB

  nm_pack_w<<<(DIM_O * DIM_H / 32) / 256, 256, 0, stream>>>(w, wpack, widx);
  convert_x<<<((size_t)DIM_M * DIM_H / 8) / 256, 256, 0, stream>>>(x, xbf);

  dim3 grid(DIM_O / 128, DIM_M / 128);   // 16 x 128 workgroups
  gemm_bf16_swmmac<<<grid, 256, 0, stream>>>(xbf, wpack, widx, bias, out);
}